// DependencyMerge_38010460569642
// MI455X (gfx1250) — compile-verified
//
#include <hip/hip_runtime.h>
#include <math.h>

#define B_   16
#define K_   64
#define C_   512
#define N0_  4096
#define N_   (K_ + N0_)      /* 4160 */
#define P_   32
#define NSEG (B_ * K_)       /* 1024 */
#define BN   (B_ * N_)       /* 66560 */

typedef float v2f __attribute__((ext_vector_type(2)));
typedef float v8f __attribute__((ext_vector_type(8)));

__device__ __forceinline__ const float* row_ptr(const float* __restrict__ proto,
                                                const float* __restrict__ x0,
                                                int b, int n) {
  return (n < K_) ? proto + ((size_t)(b * K_ + n)) * C_
                  : x0 + ((size_t)b * N0_ + (size_t)(n - K_)) * C_;
}

// ---------------- k0: seed segment-sum accumulators with the reference 1e-6 ----
__global__ void k0_init(float* __restrict__ all_c, float* __restrict__ all_s) {
  int i = blockIdx.x * blockDim.x + threadIdx.x;
  if (i < NSEG) { all_c[i] = 1e-6f; all_s[i] = 1e-6f; }
}

// ---------------- k1: per-row ||x||^2 and sigmoid(x . W + b) -------------------
__global__ __launch_bounds__(256) void k1_rowstats(
    const float* __restrict__ proto, const float* __restrict__ x0,
    const float* __restrict__ W, const float* __restrict__ bias,
    float* __restrict__ ssq, float* __restrict__ sw) {
  int lane = threadIdx.x & 31;
  int row  = blockIdx.x * 8 + (threadIdx.x >> 5);   // BN rows, 8 waves/block
  int b = row / N_, n = row % N_;
  const float* xr = row_ptr(proto, x0, b, n);
  float sq = 0.f, dt = 0.f;
#pragma unroll
  for (int i = 0; i < 4; ++i) {
    int c = lane * 4 + i * 128;
    float4 v = *(const float4*)(xr + c);
    float4 w = *(const float4*)(W + c);
    sq += v.x * v.x + v.y * v.y + v.z * v.z + v.w * v.w;
    dt += v.x * w.x + v.y * w.y + v.z * w.z + v.w * w.w;
  }
  for (int o = 16; o > 0; o >>= 1) {
    sq += __shfl_xor(sq, o, 32);
    dt += __shfl_xor(dt, o, 32);
  }
  if (lane == 0) {
    ssq[row] = sq;
    float z = dt + bias[0];
    sw[row] = 1.f / (1.f + expf(-z));
  }
}

// ---------------- k2: WMMA distance GEMM + argmin + class weights --------------
// block = 256 threads (8 waves). wave wid: kt = wid&3 (16-k tile), nsub = wid>>2
// (16-n tile). Each wave owns one 16x16 accumulator over full C=512.
__global__ __launch_bounds__(256) void k2_assign(
    const float* __restrict__ proto, const float* __restrict__ x0,
    const float* __restrict__ ssq, const float* __restrict__ sw,
    float* __restrict__ csel, int* __restrict__ idxc,
    float* __restrict__ all_c, float* __restrict__ all_s,
    float* __restrict__ out_idx) {
  __shared__ float Plds[64 * 68];      // [k][c] chunk, padded stride 68
  __shared__ float Xlds[32 * 68];      // [n][c] chunk, padded stride 68
  __shared__ float red_min[8][32];
  __shared__ int   red_idx[8][32];
  __shared__ float red_e[8][32];

  const int ntiles = N_ / 32;          // 130
  int b  = blockIdx.x / ntiles;
  int n0 = (blockIdx.x % ntiles) * 32;
  int t = threadIdx.x;
  int lane = t & 31, wid = t >> 5;
  int kt = wid & 3, nsub = wid >> 2;
  int half = lane >> 4, lm = lane & 15;

  v8f acc = {};

  for (int cc0 = 0; cc0 < C_; cc0 += 64) {
    { // stage 64x64 prototype chunk (each thread: 16 floats, coalesced)
      int r = t >> 2, cb = (t & 3) * 16;
      const float* src = proto + ((size_t)(b * K_ + r)) * C_ + cc0 + cb;
      float* dst = &Plds[r * 68 + cb];
#pragma unroll
      for (int i = 0; i < 4; ++i)
        *(float4*)(dst + 4 * i) = *(const float4*)(src + 4 * i);
    }
    { // stage 32x64 x chunk (each thread: 8 floats, coalesced)
      int r = t >> 3, cb = (t & 7) * 8;
      const float* src = row_ptr(proto, x0, b, n0 + r) + cc0 + cb;
      float* dst = &Xlds[r * 68 + cb];
#pragma unroll
      for (int i = 0; i < 2; ++i)
        *(float4*)(dst + 4 * i) = *(const float4*)(src + 4 * i);
    }
    __syncthreads();

#pragma unroll
    for (int c4 = 0; c4 < 64; c4 += 4) {
      // A 16x4 f32: VGPR0 = K{0|2}, VGPR1 = K{1|3}; lane half selects K pair.
      float2 av = *(const float2*)&Plds[(kt * 16 + lm) * 68 + c4 + 2 * half];
      // B 4x16 f32: mirrored layout, lane index = column n.
      float2 bv = *(const float2*)&Xlds[(nsub * 16 + lm) * 68 + c4 + 2 * half];
      v2f a;  a[0] = av.x;  a[1] = av.y;
      v2f bb; bb[0] = bv.x; bb[1] = bv.y;
      acc = __builtin_amdgcn_wmma_f32_16x16x4_f32(
          false, a, false, bb, (short)0, acc, false, false);
    }
    __syncthreads();
  }

  // Per-lane: 8 k values (k = kt*16 + half*8 + r) for column n = n0+nsub*16+lm.
  int nloc = nsub * 16 + lm;
  int n = n0 + nloc;
  float sx = ssq[b * N_ + n];
  const float inv_sqrtC = 0.044194173824159216f;   // 1/sqrt(512)
  float bd = 3.4e38f; int bk = 0; float es = 0.f;
#pragma unroll
  for (int r = 0; r < 8; ++r) {
    int k = kt * 16 + half * 8 + r;                // ascending k within slot
    float sp = ssq[b * N_ + k];                    // prototype rows are n<K
    float d2 = sp + sx - 2.f * acc[r];
    float dist = sqrtf(fmaxf(d2, 0.f)) * inv_sqrtC;
    es += expf(-dist);
    if (dist < bd) { bd = dist; bk = k; }
  }
  int slot = kt * 2 + half;                        // slot s -> k base 8*s
  red_min[slot][nloc] = bd;
  red_idx[slot][nloc] = bk;
  red_e[slot][nloc]   = es;
  __syncthreads();

  if (t < 32) {                                    // one thread per column
    float m = 3.4e38f; int ki = 0; float e0 = 0.f, e1 = 0.f;
#pragma unroll
    for (int s = 0; s < 8; ++s) {                  // ascending k: first-min ties
      float mv = red_min[s][t];
      int   kv = red_idx[s][t];
      float ev = red_e[s][t];
      if (mv < m) { m = mv; ki = kv; }
      if (s < 4) e0 += ev; else e1 += ev;          // class 0: k<32
    }
    int gi = b * N_ + n0 + t;
    float cw0 = e0 * (1.f / P_), cw1 = e1 * (1.f / P_);
    float den = cw0 + cw1 + 1e-6f;
    float cs = ((ki >> 5) ? cw1 : cw0) / den;
    csel[gi] = cs;
    idxc[gi] = ki;
    out_idx[gi] = (float)ki;
    atomicAdd(&all_c[b * K_ + ki], cs);
    atomicAdd(&all_s[b * K_ + ki], sw[gi]);
  }
}

// ---------------- k3: deterministic gather segment-sum into merged -------------
__global__ __launch_bounds__(256) void k3_merge(
    const float* __restrict__ proto, const float* __restrict__ x0,
    const float* __restrict__ csel, const float* __restrict__ sw,
    const int* __restrict__ idxc,
    const float* __restrict__ all_c, const float* __restrict__ all_s,
    float* __restrict__ merged) {
  int b = blockIdx.x / K_, k = blockIdx.x % K_;
  float inv_ac = 0.5f / all_c[b * K_ + k];
  float inv_as = 0.5f / all_s[b * K_ + k];
  int t = threadIdx.x;
  float a0 = 0.f, a1 = 0.f;
  for (int n = 0; n < N_; ++n) {
    int gi = b * N_ + n;
    if (idxc[gi] == k) {
      float w = csel[gi] * inv_ac + sw[gi] * inv_as;
      const float* xr = row_ptr(proto, x0, b, n);
      a0 += xr[t] * w;
      a1 += xr[t + 256] * w;
    }
  }
  size_t obase = ((size_t)(b * K_ + k)) * C_;
  merged[obase + t]       = a0;
  merged[obase + t + 256] = a1;
}

extern "C" void kernel_launch(void* const* d_in, const int* in_sizes, int n_in,
                              void* d_out, int out_size, void* d_ws, size_t ws_size,
                              hipStream_t stream) {
  const float* proto = (const float*)d_in[0];   // [B,K,C]
  const float* x0    = (const float*)d_in[1];   // [B,N0,C]
  const float* W     = (const float*)d_in[2];   // [C,1]
  const float* bias  = (const float*)d_in[3];   // [1]

  float* merged  = (float*)d_out;               // [B,K,C]
  float* out_idx = merged + (size_t)NSEG * C_;  // [B,N] as float

  float* ws    = (float*)d_ws;
  float* ssq   = ws;                 // BN
  float* sw    = ws + BN;            // BN
  float* csel  = ws + 2 * BN;        // BN
  int*   idxc  = (int*)(ws + 3 * BN);// BN
  float* all_c = ws + 4 * BN;        // NSEG
  float* all_s = ws + 4 * BN + NSEG; // NSEG

  k0_init<<<(NSEG + 255) / 256, 256, 0, stream>>>(all_c, all_s);
  k1_rowstats<<<BN / 8, 256, 0, stream>>>(proto, x0, W, bias, ssq, sw);
  k2_assign<<<B_ * (N_ / 32), 256, 0, stream>>>(proto, x0, ssq, sw, csel, idxc,
                                                all_c, all_s, out_idx);
  k3_merge<<<NSEG, 256, 0, stream>>>(proto, x0, csel, sw, idxc, all_c, all_s,
                                     merged);
}